// Resample_48498770706999
// MI455X (gfx1250) — compile-verified
//
#include <hip/hip_runtime.h>

// 2x upsample + separable [k0,k1,k2,k3] filter (wrap W, edge H), f32.
// Input  (8,128,64,512)  -> Output (8,128,128,1024).
// Memory-bound: 640 MiB moved, ~28 us floor @ 23.3 TB/s on MI455X.
// Staging: CDNA5 async global->LDS gather (ASYNCcnt). Output: b128 NT streams.

typedef float v4f __attribute__((ext_vector_type(4)));

#define BLOCK_THREADS 256
#define TILE_H 16
#define TILE_W 64
#define LDS_H (TILE_H + 2)
#define LDS_W (TILE_W + 2)
#define LDS_N (LDS_H * LDS_W)

static constexpr int IN_H = 64, IN_W = 512;
static constexpr int OUT_H = 128, OUT_W = 1024;

typedef __attribute__((address_space(1))) int  gint_t;
typedef __attribute__((address_space(3))) int  lint_t;

// ---- CDNA5 async global->LDS copy (gather form: per-lane LDS + global addr) ----
__device__ __forceinline__ void async_copy_b32(const float* gsrc, float* lds_dst) {
#if __has_builtin(__builtin_amdgcn_global_load_async_to_lds_b32)
  __builtin_amdgcn_global_load_async_to_lds_b32(
      (gint_t*)const_cast<float*>(gsrc),
      (lint_t*)lds_dst,
      0, 0);
#else
  unsigned lds_off =
      (unsigned)(unsigned long long)(__attribute__((address_space(3))) float*)lds_dst;
  asm volatile("global_load_async_to_lds_b32 %0, %1, off"
               :: "v"(lds_off), "v"(gsrc) : "memory");
#endif
}

__device__ __forceinline__ void async_wait0() {
#if __has_builtin(__builtin_amdgcn_s_wait_asynccnt)
  __builtin_amdgcn_s_wait_asynccnt(0);
#else
  asm volatile("s_wait_asynccnt 0" ::: "memory");
#endif
}

__global__ __launch_bounds__(BLOCK_THREADS)
void Resample_48498770706999_kernel(const float* __restrict__ src_all,
                                    const float* __restrict__ kern,
                                    float* __restrict__ dst_all) {
  __shared__ float tile[LDS_N];

  const int plane = blockIdx.z;                // b*C + c  (0..1023)
  const int row0  = blockIdx.y * TILE_H;       // input row base of tile
  const int col0  = blockIdx.x * TILE_W;       // input col base of tile
  const int tid   = threadIdx.x;

  const float* __restrict__ src = src_all + (size_t)plane * (IN_H * IN_W);
  float* __restrict__       dst = dst_all + (size_t)plane * (OUT_H * OUT_W);

  // ---- stage (TILE_H+2) x (TILE_W+2) halo tile into LDS via async gather ----
  // tile[r][c] = in[clamp(row0 + r - 1), (col0 + c - 1) mod 512]
  for (int e = tid; e < LDS_N; e += BLOCK_THREADS) {
    int lr = e / LDS_W;
    int lc = e - lr * LDS_W;
    int gr = row0 + lr - 1;
    gr = gr < 0 ? 0 : (gr >= IN_H ? IN_H - 1 : gr);      // edge clamp (H)
    int gc = (col0 + lc - 1) & (IN_W - 1);               // wrap (W, pow2)
    async_copy_b32(src + gr * IN_W + gc, &tile[e]);
  }
  async_wait0();        // own wave's async loads landed in LDS
  __syncthreads();      // everyone else's too

  const float k0 = kern[0], k1 = kern[1], k2 = kern[2], k3 = kern[3];

  // Each thread: 2 iterations, each producing a 2x4 output tile (two b128 NT
  // stores) from a 3x4 LDS neighborhood covering input cols n-1..n+2.
#pragma unroll
  for (int it = 0; it < (TILE_H * TILE_W / 2) / BLOCK_THREADS; ++it) {
    const int p  = tid + it * BLOCK_THREADS;
    const int lr = p >> 5;               // pair-row   (0..15)
    const int lc = (p & 31) << 1;        // even input col within tile (0..62)

    const float* t0 = &tile[lr * LDS_W + lc];  // row m-1, starting at col n-1
    const float* t1 = t0 + LDS_W;              // row m
    const float* t2 = t1 + LDS_W;              // row m+1

    const float a0 = t0[0], a1 = t0[1], a2 = t0[2], a3 = t0[3];
    const float b0 = t1[0], b1 = t1[1], b2 = t1[2], b3 = t1[3];
    const float c0 = t2[0], c1 = t2[1], c2 = t2[2], c3 = t2[3];

    // horizontal polyphase combos per row (cols n and n+1, even/odd phases)
    const float aE0 = k0 * a0 + k2 * a1,  aO0 = k1 * a1 + k3 * a2;
    const float aE1 = k0 * a1 + k2 * a2,  aO1 = k1 * a2 + k3 * a3;
    const float bE0 = k0 * b0 + k2 * b1,  bO0 = k1 * b1 + k3 * b2;
    const float bE1 = k0 * b1 + k2 * b2,  bO1 = k1 * b2 + k3 * b3;
    const float cE0 = k0 * c0 + k2 * c1,  cO0 = k1 * c1 + k3 * c2;
    const float cE1 = k0 * c1 + k2 * c2,  cO1 = k1 * c2 + k3 * c3;

    // vertical polyphase
    v4f r_even = { k0 * aE0 + k2 * bE0,     // y = 2m,   x = 2n
                   k0 * aO0 + k2 * bO0,     //           x = 2n+1
                   k0 * aE1 + k2 * bE1,     //           x = 2n+2
                   k0 * aO1 + k2 * bO1 };   //           x = 2n+3
    v4f r_odd  = { k1 * bE0 + k3 * cE0,     // y = 2m+1
                   k1 * bO0 + k3 * cO0,
                   k1 * bE1 + k3 * cE1,
                   k1 * bO1 + k3 * cO1 };

    const int m = row0 + lr;
    const int n = col0 + lc;                 // even -> 2n multiple of 4 -> 16B aligned
    v4f* o0 = (v4f*)(dst + (size_t)(2 * m)     * OUT_W + 2 * n);
    v4f* o1 = (v4f*)(dst + (size_t)(2 * m + 1) * OUT_W + 2 * n);
    __builtin_nontemporal_store(r_even, o0); // 512 MiB stream: bypass caches
    __builtin_nontemporal_store(r_odd,  o1);
  }
}

extern "C" void kernel_launch(void* const* d_in, const int* in_sizes, int n_in,
                              void* d_out, int out_size, void* d_ws, size_t ws_size,
                              hipStream_t stream) {
  const float* h    = (const float*)d_in[0];   // (8,128,64,512) f32
  const float* kern = (const float*)d_in[1];   // 4 filter taps f32
  float* out        = (float*)d_out;           // (8,128,128,1024) f32

  const int planes = in_sizes[0] / (IN_H * IN_W);     // 8*128 = 1024
  dim3 grid(IN_W / TILE_W, IN_H / TILE_H, planes);    // (8, 4, 1024)
  Resample_48498770706999_kernel<<<grid, BLOCK_THREADS, 0, stream>>>(h, kern, out);
}